// DispAgg_34359738499
// MI455X (gfx1250) — compile-verified
//
#include <hip/hip_runtime.h>

// ---------------- geometry ----------------
#define C_IN   32
#define D0     64
#define H0     96
#define W0     192
#define HW0    (H0*W0)            // 18432
#define CS0    (D0*H0*W0)         // x channel stride 1179648
#define DD     193                // MAXDISP+1
#define HH     288
#define WW     576
#define HWOUT  (HH*WW)            // 165888
// padded volume: pd = d+1 (d in 0..192 -> pd 1..193; pd 0,194,195 zero halo)
//                ph = h+2, pw = w+4
#define DP     196
#define HP     292
#define WP     584
#define HPWP   (HP*WP)            // 170528
#define VOLP   ((size_t)DP*(size_t)HPWP)   // 33,423,488 floats

typedef __attribute__((ext_vector_type(2))) float        v2f;
typedef __attribute__((ext_vector_type(8))) float        v8f;
typedef __attribute__((ext_vector_type(4))) unsigned int v4u;
typedef __attribute__((ext_vector_type(8))) int          v8i;
typedef __attribute__((ext_vector_type(4))) int          v4i;

#if __has_builtin(__builtin_amdgcn_tensor_load_to_lds) && \
    __has_builtin(__builtin_amdgcn_s_wait_tensorcnt)
#define USE_TDM 1
#endif

// ---------------- zero fill (halos + buffers) ----------------
__global__ void k_zero4(float4* __restrict__ p, long n4) {
    long i = (long)blockIdx.x * blockDim.x + threadIdx.x;
    long stride = (long)gridDim.x * blockDim.x;
    float4 z = make_float4(0.f, 0.f, 0.f, 0.f);
    for (; i < n4; i += stride) p[i] = z;
}

// ---------------- conv3d: 32ch -> 1ch, 3x3x3, pad 1 ----------------
// Weights (864 floats) staged into LDS via the Tensor Data Mover.
__global__ void k_conv3d(const float* __restrict__ x, const float* __restrict__ cw,
                         float* __restrict__ cost) {
    __shared__ float lw[C_IN * 27];
#ifdef USE_TDM
    if (threadIdx.x < 32) {      // wave-uniform: only wave 0 issues the DMA
        unsigned long long ga = (unsigned long long)(uintptr_t)cw;
        v4u g0;
        g0.x = 1u;                                             // count=1, no gather
        g0.y = 0u;                                             // lds_addr: lw is sole LDS object -> 0
        g0.z = (unsigned)(ga & 0xffffffffu);                   // global_addr[31:0]
        g0.w = (unsigned)((ga >> 32) & 0x1ffffffu) | (2u << 30); // global_addr[56:32] | type=2
        v8i g1;
        g1[0] = (int)(2u << 16);        // wg_mask=0, data_size=2 (4 bytes)
        g1[1] = (int)((C_IN * 27u) << 16); // tensor_dim0[15:0] = 864
        g1[2] = (int)(1u << 16);        // tensor_dim0 hi=0, tensor_dim1[15:0]=1
        g1[3] = (int)((C_IN * 27u) << 16); // tensor_dim1 hi=0, tile_dim0 = 864
        g1[4] = 0;                      // tile_dim1=0 (unused), tile_dim2=0
        g1[5] = C_IN * 27;              // tensor_dim0_stride lo32
        g1[6] = 0;
        g1[7] = 0;
        v4i gz4 = {0, 0, 0, 0};         // groups 2/3 unused (<=2D)
        v8i gz8 = {0, 0, 0, 0, 0, 0, 0, 0};
        __builtin_amdgcn_tensor_load_to_lds(g0, g1, gz4, gz4, gz8, 0);
        __builtin_amdgcn_s_wait_tensorcnt(0);
    }
    __syncthreads();
    const float* lwp = lw;
    asm volatile("" : "+v"(lwp) :: "memory");   // TDM wrote LDS behind compiler's back
#else
    for (int i = threadIdx.x; i < C_IN * 27; i += blockDim.x) lw[i] = cw[i];
    __syncthreads();
    const float* lwp = lw;
#endif
    int gid = blockIdx.x * 256 + threadIdx.x;   // < 1179648 (exact)
    int w = gid % W0;
    int h = (gid / W0) % H0;
    int d = gid / (W0 * H0);
    float acc = 0.f;
    for (int c = 0; c < C_IN; ++c) {
        const float* xb = x + (size_t)c * CS0;
        const float* wb = lwp + c * 27;
        #pragma unroll
        for (int kd = 0; kd < 3; ++kd) {
            int zd = d + kd - 1;
            if (zd < 0 || zd >= D0) continue;
            #pragma unroll
            for (int kh = 0; kh < 3; ++kh) {
                int zh = h + kh - 1;
                if (zh < 0 || zh >= H0) continue;
                const float* row = xb + (size_t)zd * HW0 + zh * W0;
                #pragma unroll
                for (int kw = 0; kw < 3; ++kw) {
                    int zw = w + kw - 1;
                    float v = (zw >= 0 && zw < W0) ? row[zw] : 0.f;
                    acc = fmaf(wb[kd * 9 + kh * 3 + kw], v, acc);
                }
            }
        }
    }
    cost[gid] = acc;
}

// ---------------- fused trilinear resize (64,96,192)->(193,288,576) ----------------
__device__ __forceinline__ void lerp_idx(int o, float scale, int inmax,
                                         int& lo, int& hi, float& wt) {
    float src = ((float)o + 0.5f) * scale - 0.5f;
    src = fminf(fmaxf(src, 0.f), (float)inmax);
    float f = floorf(src);
    lo = (int)f;
    hi = min(lo + 1, inmax);
    wt = src - f;
}

__global__ void k_resize(const float* __restrict__ cost, float* __restrict__ cP) {
    int gid = blockIdx.x * 256 + threadIdx.x;   // < 193*165888 (exact)
    int w = gid % WW;
    int h = (gid / WW) % HH;
    int d = gid / (WW * HH);
    int dlo, dhi, hlo, hhi, wlo, whi;
    float wd, wh, ww;
    lerp_idx(d, 64.f / 193.f, D0 - 1, dlo, dhi, wd);
    lerp_idx(h, 96.f / 288.f, H0 - 1, hlo, hhi, wh);
    lerp_idx(w, 192.f / 576.f, W0 - 1, wlo, whi, ww);
    const float* base_lo = cost + (size_t)dlo * HW0;
    const float* base_hi = cost + (size_t)dhi * HW0;
    float v000 = base_lo[hlo * W0 + wlo], v001 = base_lo[hlo * W0 + whi];
    float v010 = base_lo[hhi * W0 + wlo], v011 = base_lo[hhi * W0 + whi];
    float v100 = base_hi[hlo * W0 + wlo], v101 = base_hi[hlo * W0 + whi];
    float v110 = base_hi[hhi * W0 + wlo], v111 = base_hi[hhi * W0 + whi];
    float v00 = v000 + (v001 - v000) * ww;
    float v01 = v010 + (v011 - v010) * ww;
    float v10 = v100 + (v101 - v100) * ww;
    float v11 = v110 + (v111 - v110) * ww;
    float v0 = v00 + (v01 - v00) * wh;
    float v1 = v10 + (v11 - v10) * wh;
    float v = v0 + (v1 - v0) * wd;
    cP[(size_t)(d + 1) * HPWP + (size_t)(h + 2) * WP + (w + 4)] = v;
}

// ---------------- L1 normalize 75-channel guidance over channel axis ----------------
__global__ void k_l1norm75(const float* __restrict__ g, float* __restrict__ gn) {
    int pix = blockIdx.x * 256 + threadIdx.x;   // < 165888 (exact)
    float s = 0.f;
    for (int c = 0; c < 75; ++c) s += fabsf(g[(size_t)c * HWOUT + pix]);
    float inv = 1.f / fmaxf(s, 1e-12f);
    for (int c = 0; c < 75; ++c)
        gn[(size_t)c * HWOUT + pix] = g[(size_t)c * HWOUT + pix] * inv;
}

// ---------------- one LGA pass on the padded volume ----------------
// out[d] = sum_{dd,i,j} g[dd*25+i*5+j] * in[d+dd-1, h+i-2, w+j-2]
// Slice s contributes S0(s)->out[s+1], S1(s)->out[s], S2(s)->out[s-1].
__global__ void k_lga(const float* __restrict__ in, const float* __restrict__ gn,
                      float* __restrict__ out) {
    int pix = blockIdx.x * 256 + threadIdx.x;   // < 165888 (exact)
    int w = pix % WW;
    int h = pix / WW;
    float gw[75];
    #pragma unroll
    for (int c = 0; c < 75; ++c) gw[c] = gn[(size_t)c * HWOUT + pix];
    // slice-local window base: rows h..h+4 (= ph-2..ph+2), cols w+2..w+6 (= pw-2..pw+2)
    const float* base = in + (size_t)h * WP + (w + 2);
    float* obase = out + (size_t)(h + 2) * WP + (w + 4);
    float accA = 0.f, accB = 0.f;   // accA: partial out[s-1], accB: partial out[s]
    for (int s = 0; s < DD; ++s) {
        const float* sp = base + (size_t)(s + 1) * HPWP;
        __builtin_prefetch(sp + 2 * (size_t)HPWP, 0, 1);   // gfx1250 global_prefetch_b8
        float S0 = 0.f, S1 = 0.f, S2 = 0.f;
        #pragma unroll
        for (int r = 0; r < 5; ++r) {
            const float* rp = sp + (size_t)r * WP;
            float v0 = rp[0], v1 = rp[1], v2 = rp[2], v3 = rp[3], v4 = rp[4];
            S0 = fmaf(gw[r * 5 + 0], v0, S0); S0 = fmaf(gw[r * 5 + 1], v1, S0);
            S0 = fmaf(gw[r * 5 + 2], v2, S0); S0 = fmaf(gw[r * 5 + 3], v3, S0);
            S0 = fmaf(gw[r * 5 + 4], v4, S0);
            S1 = fmaf(gw[25 + r * 5 + 0], v0, S1); S1 = fmaf(gw[25 + r * 5 + 1], v1, S1);
            S1 = fmaf(gw[25 + r * 5 + 2], v2, S1); S1 = fmaf(gw[25 + r * 5 + 3], v3, S1);
            S1 = fmaf(gw[25 + r * 5 + 4], v4, S1);
            S2 = fmaf(gw[50 + r * 5 + 0], v0, S2); S2 = fmaf(gw[50 + r * 5 + 1], v1, S2);
            S2 = fmaf(gw[50 + r * 5 + 2], v2, S2); S2 = fmaf(gw[50 + r * 5 + 3], v3, S2);
            S2 = fmaf(gw[50 + r * 5 + 4], v4, S2);
        }
        float ov = accA + S2;              // completes out[s-1]
        if (s > 0) obase[(size_t)s * HPWP] = ov;   // pd = s  <->  d = s-1
        accA = accB + S1;
        accB = S0;
    }
    obase[(size_t)DD * HPWP] = accA;       // out[192] = S0(191)+S1(192) (S2(193)=0)
}

// ---------------- softmax(-c) over depth, per pixel ----------------
__global__ void k_softmax(const float* __restrict__ cP, float* __restrict__ pP) {
    int pix = blockIdx.x * 256 + threadIdx.x;   // < 165888 (exact)
    int w = pix % WW;
    int h = pix / WW;
    size_t off = (size_t)HPWP + (size_t)(h + 2) * WP + (w + 4);   // pd = 1 (d = 0)
    const float* c = cP + off;
    float* p = pP + off;
    float mx = -3.4e38f;
    for (int d = 0; d < DD; ++d) mx = fmaxf(mx, -c[(size_t)d * HPWP]);
    float s = 0.f;
    for (int d = 0; d < DD; ++d) {
        float e = __expf(-c[(size_t)d * HPWP] - mx);
        p[(size_t)d * HPWP] = e;
        s += e;
    }
    float inv = 1.f / s;
    for (int d = 0; d < DD; ++d) p[(size_t)d * HPWP] *= inv;
}

// ---------------- final: L1-normalize + expectation via WMMA ----------------
// Per wave: 16 pixels. D(16x16) += A(16x4) x B(4x16) over 49 K-chunks (196 padded
// depth slices; halos are zero). A row0 = disparity values (pd-1), row1 = ones.
// D row0 = sum d*p, row1 = sum p (p >= 0 so |p| == p).
// A is loop-carried: row-0 coefficients advance by +4 per K-chunk.
__global__ void k_finalize(const float* __restrict__ pP, float* __restrict__ out) {
    int lane = threadIdx.x & 31;
    int wave = blockIdx.x * (blockDim.x >> 5) + (threadIdx.x >> 5);
    int pixBase = wave * 16;                 // 165888 = 10368 waves * 16 (exact)
    int m  = lane & 15;                      // A row / B col / output pixel slot
    int kb = (lane >> 4) << 1;               // K pair base: 0 (lanes 0-15) or 2 (16-31)
    int pix = pixBase + m;                   // 16-pixel group never wraps a row (576%16==0)
    int w = pix % WW;
    int h = pix / WW;
    const float* col = pP + (size_t)(h + 2) * WP + (w + 4);
    v8f acc = {0.f, 0.f, 0.f, 0.f, 0.f, 0.f, 0.f, 0.f};
    v2f a;
    a.x = (m == 0) ? (float)(kb - 1) : ((m == 1) ? 1.f : 0.f);  // K = kb
    a.y = (m == 0) ? (float)(kb)     : ((m == 1) ? 1.f : 0.f);  // K = kb+1
    float astep = (m == 0) ? 4.f : 0.f;
    for (int it = 0; it < 49; ++it) {
        int pd0 = it * 4;
        v2f b;
        b.x = col[(size_t)(pd0 + kb)     * HPWP];
        b.y = col[(size_t)(pd0 + kb + 1) * HPWP];
        acc = __builtin_amdgcn_wmma_f32_16x16x4_f32(
            false, a, false, b, (short)0, acc, false, false);
        a.x += astep;
        a.y += astep;
    }
    if (lane < 16) {
        float num = acc[0];                  // D row 0, N = lane
        float den = acc[1];                  // D row 1, N = lane
        out[pixBase + lane] = num / fmaxf(den, 1e-12f);
    }
}

// ---------------- launch ----------------
extern "C" void kernel_launch(void* const* d_in, const int* in_sizes, int n_in,
                              void* d_out, int out_size, void* d_ws, size_t ws_size,
                              hipStream_t stream) {
    (void)in_sizes; (void)n_in; (void)out_size; (void)ws_size;
    const float* x   = (const float*)d_in[0];
    const float* lg1 = (const float*)d_in[1];
    const float* lg2 = (const float*)d_in[2];
    const float* cw  = (const float*)d_in[3];
    float* ws = (float*)d_ws;
    float* cP   = ws;                                   // padded volume A
    float* tP   = ws + VOLP;                            // padded volume B
    float* g1   = ws + 2 * VOLP;                        // 75*165888
    float* g2   = g1 + (size_t)75 * HWOUT;
    float* cost = g2 + (size_t)75 * HWOUT;              // 1179648
    float* out  = (float*)d_out;

    long n4 = (long)(2 * VOLP) / 4;
    k_zero4   <<<8192, 256, 0, stream>>>((float4*)ws, n4);
    k_conv3d  <<<4608, 256, 0, stream>>>(x, cw, cost);          // 1179648 / 256
    k_resize  <<<125064, 256, 0, stream>>>(cost, cP);           // 193*165888 / 256
    k_l1norm75<<<648, 256, 0, stream>>>(lg1, g1);               // 165888 / 256
    k_l1norm75<<<648, 256, 0, stream>>>(lg2, g2);
    k_lga     <<<648, 256, 0, stream>>>(cP, g1, tP);
    k_lga     <<<648, 256, 0, stream>>>(tP, g1, cP);
    k_softmax <<<648, 256, 0, stream>>>(cP, tP);
    k_lga     <<<648, 256, 0, stream>>>(tP, g2, cP);
    k_lga     <<<648, 256, 0, stream>>>(cP, g2, tP);
    k_finalize<<<1296, 256, 0, stream>>>(tP, out);              // 10368 waves * 16 px
}